// Criterion_48773648613659
// MI455X (gfx1250) — compile-verified
//
#include <hip/hip_runtime.h>

typedef float v2f __attribute__((ext_vector_type(2)));
typedef float v8f __attribute__((ext_vector_type(8)));

#define NODE_TYPE_HANDLE_V 3
#define CORR_EPS 0.5f
#define PEN_EPS 0.5f
#define PINNED_W 10.0f
#define WEIGHT_MAX_V 100000.0f
#define STEP_MAX_V 1000.0f

// ---------------------------------------------------------------------------
// Phase 1a: per-node table. node_prev/pred[i] = {x,y,z, pinned?1:0}
// Also zeroes the output accumulator (thread 0).
// ---------------------------------------------------------------------------
__global__ void crit_node_table(const float* __restrict__ prev_pos,
                                const float* __restrict__ pred_pos,
                                const int* __restrict__ vtype,
                                float4* __restrict__ node_prev,
                                float4* __restrict__ node_pred,
                                float* __restrict__ out,
                                int n_nodes) {
    int i = blockIdx.x * blockDim.x + threadIdx.x;
    if (i == 0) out[0] = 0.0f;
    int stride = gridDim.x * blockDim.x;
    for (; i < n_nodes; i += stride) {
        float pin = (vtype[i] == NODE_TYPE_HANDLE_V) ? 1.0f : 0.0f;
        float4 p, q;
        p.x = prev_pos[3 * i + 0];
        p.y = prev_pos[3 * i + 1];
        p.z = prev_pos[3 * i + 2];
        p.w = pin;
        q.x = pred_pos[3 * i + 0];
        q.y = pred_pos[3 * i + 1];
        q.z = pred_pos[3 * i + 2];
        q.w = pin;
        node_prev[i] = p;
        node_pred[i] = q;
    }
}

// ---------------------------------------------------------------------------
// Phase 1b: per-face table. Unit normal + centroid for prev & pred positions.
// tri_pinned (any vertex pinned) packed into f_nprev.w.
// ---------------------------------------------------------------------------
__device__ __forceinline__ void plane_of(const float* __restrict__ pos,
                                         int v0, int v1, int v2,
                                         float4& n_out, float4& c_out) {
    float ax = pos[3 * v0], ay = pos[3 * v0 + 1], az = pos[3 * v0 + 2];
    float bx = pos[3 * v1], by = pos[3 * v1 + 1], bz = pos[3 * v1 + 2];
    float cx = pos[3 * v2], cy = pos[3 * v2 + 1], cz = pos[3 * v2 + 2];
    float e1x = bx - ax, e1y = by - ay, e1z = bz - az;
    float e2x = cx - ax, e2y = cy - ay, e2z = cz - az;
    float nx = e1y * e2z - e1z * e2y;
    float ny = e1z * e2x - e1x * e2z;
    float nz = e1x * e2y - e1y * e2x;
    float inv = 1.0f / (sqrtf(nx * nx + ny * ny + nz * nz) + 1e-12f);
    n_out.x = nx * inv; n_out.y = ny * inv; n_out.z = nz * inv; n_out.w = 0.0f;
    const float third = 1.0f / 3.0f;
    c_out.x = (ax + bx + cx) * third;
    c_out.y = (ay + by + cy) * third;
    c_out.z = (az + bz + cz) * third;
    c_out.w = 0.0f;
}

__global__ void crit_face_table(const float* __restrict__ prev_pos,
                                const float* __restrict__ pred_pos,
                                const int* __restrict__ faces,
                                const int* __restrict__ vtype,
                                float4* __restrict__ f_nprev,
                                float4* __restrict__ f_cprev,
                                float4* __restrict__ f_npred,
                                float4* __restrict__ f_cpred,
                                int n_faces) {
    int i = blockIdx.x * blockDim.x + threadIdx.x;
    int stride = gridDim.x * blockDim.x;
    for (; i < n_faces; i += stride) {
        int v0 = faces[3 * i + 0];
        int v1 = faces[3 * i + 1];
        int v2 = faces[3 * i + 2];
        float4 n0, c0, n1, c1;
        plane_of(prev_pos, v0, v1, v2, n0, c0);
        plane_of(pred_pos, v0, v1, v2, n1, c1);
        bool tp = (vtype[v0] == NODE_TYPE_HANDLE_V) ||
                  (vtype[v1] == NODE_TYPE_HANDLE_V) ||
                  (vtype[v2] == NODE_TYPE_HANDLE_V);
        n0.w = tp ? 1.0f : 0.0f;  // tri_pinned rides with the prev-normal
        f_nprev[i] = n0;
        f_cprev[i] = c0;
        f_npred[i] = n1;
        f_cpred[i] = c1;
    }
}

// ---------------------------------------------------------------------------
// Phase 2: per-edge hinge penalty + reduction.
// Streams nodes_from / faces_to coalesced; all gathers hit the L2-resident
// tables built in phase 1. Wave reduction: one V_WMMA_F32_16X16X4_F32 with
// A = ones collapses 32 lane-partials to 16 column sums, then 4 shfl_xor
// steps; one atomicAdd per wave.
// ---------------------------------------------------------------------------
__global__ void crit_edge_loss(const int* __restrict__ nodes_from,
                               const int* __restrict__ faces_to,
                               const float4* __restrict__ node_prev,
                               const float4* __restrict__ node_pred,
                               const float4* __restrict__ f_nprev,
                               const float4* __restrict__ f_cprev,
                               const float4* __restrict__ f_npred,
                               const float4* __restrict__ f_cpred,
                               const int* __restrict__ step_p,
                               const int* __restrict__ iter_p,
                               float* __restrict__ out,
                               int n_edges) {
    float acc = 0.0f;
    int stride = gridDim.x * blockDim.x;
    for (int e = blockIdx.x * blockDim.x + threadIdx.x; e < n_edges; e += stride) {
        int nf = nodes_from[e];
        int ft = faces_to[e];
        float4 P  = node_prev[nf];
        float4 FN = f_nprev[ft];
        float4 FC = f_cprev[ft];
        float d_prev = (P.x - FC.x) * FN.x + (P.y - FC.y) * FN.y + (P.z - FC.z) * FN.z;
        if (fabsf(d_prev) < CORR_EPS) {
            float4 Q  = node_pred[nf];
            float4 GN = f_npred[ft];
            float4 GC = f_cpred[ft];
            float d_curr = (Q.x - GC.x) * GN.x + (Q.y - GC.y) * GN.y + (Q.z - GC.z) * GN.z;
            float s = (d_prev > 0.0f) ? 1.0f : ((d_prev < 0.0f) ? -1.0f : 0.0f);
            d_curr *= s;  // SIGN == 1.0
            float interp = fmaxf(PEN_EPS - d_curr, 0.0f);
            bool full_pinned = (P.w != 0.0f) || (FN.w != 0.0f);
            if (full_pinned) interp *= PINNED_W;
            acc += interp * interp * interp;
        }
    }

    // Cross-lane reduction via WMMA (EXEC all-ones here: grid-stride loop
    // reconverged, no early returns). A = ones(16x4), B row K holds lane
    // partials -> D[m][n] = partial(n) + partial(n+16); result is invariant
    // to the exact K-row assignment since we sum all columns afterwards.
    v2f a; a.x = 1.0f; a.y = 1.0f;
    v2f b; b.x = acc;  b.y = 0.0f;
    v8f c = {};
    c = __builtin_amdgcn_wmma_f32_16x16x4_f32(
        /*neg_a=*/false, a, /*neg_b=*/false, b,
        /*c_mod=*/(short)0, c, /*reuse_a=*/false, /*reuse_b=*/false);
    float colsum = c[0];
    #pragma unroll
    for (int off = 1; off < 16; off <<= 1)
        colsum += __shfl_xor(colsum, off, 32);

    if ((threadIdx.x & 31) == 0) {
        int it = iter_p[0];
        int st = step_p[0];
        float prog = fminf(fmaxf((float)it, 0.0f) / STEP_MAX_V, 1.0f);
        float w = WEIGHT_MAX_V * prog;   // WEIGHT_START = 0
        if (st == 0) w = 0.0f;
        if (w != 0.0f) atomicAdd(out, colsum * w);  // B == 1
    }
}

// ---------------------------------------------------------------------------
// Launcher. Input order (setup_inputs): prev_pos, pred_pos, faces,
// vertex_type, nodes_from, faces_to, step, iter_num.
// ---------------------------------------------------------------------------
extern "C" void kernel_launch(void* const* d_in, const int* in_sizes, int n_in,
                              void* d_out, int out_size, void* d_ws, size_t ws_size,
                              hipStream_t stream) {
    const float* prev_pos  = (const float*)d_in[0];
    const float* pred_pos  = (const float*)d_in[1];
    const int*   faces     = (const int*)d_in[2];
    const int*   vtype     = (const int*)d_in[3];
    const int*   nodes_from= (const int*)d_in[4];
    const int*   faces_to  = (const int*)d_in[5];
    const int*   step_p    = (const int*)d_in[6];
    const int*   iter_p    = (const int*)d_in[7];
    float* out = (float*)d_out;

    const int n_nodes = in_sizes[0] / 3;
    const int n_faces = in_sizes[2] / 3;
    const int n_edges = in_sizes[4];

    // Workspace layout (all 16B aligned):
    char* ws = (char*)d_ws;
    size_t off = 0;
    float4* node_prev = (float4*)(ws + off); off += (size_t)n_nodes * sizeof(float4);
    float4* node_pred = (float4*)(ws + off); off += (size_t)n_nodes * sizeof(float4);
    float4* f_nprev   = (float4*)(ws + off); off += (size_t)n_faces * sizeof(float4);
    float4* f_cprev   = (float4*)(ws + off); off += (size_t)n_faces * sizeof(float4);
    float4* f_npred   = (float4*)(ws + off); off += (size_t)n_faces * sizeof(float4);
    float4* f_cpred   = (float4*)(ws + off); off += (size_t)n_faces * sizeof(float4);
    (void)ws_size; (void)n_in; (void)out_size;

    const int T = 256;
    int nb_nodes = (n_nodes + T - 1) / T;
    int nb_faces = (n_faces + T - 1) / T;
    int nb_edges = (n_edges + T - 1) / T;
    if (nb_edges > 3072) nb_edges = 3072;

    crit_node_table<<<nb_nodes, T, 0, stream>>>(prev_pos, pred_pos, vtype,
                                                node_prev, node_pred, out, n_nodes);
    crit_face_table<<<nb_faces, T, 0, stream>>>(prev_pos, pred_pos, faces, vtype,
                                                f_nprev, f_cprev, f_npred, f_cpred,
                                                n_faces);
    crit_edge_loss<<<nb_edges, T, 0, stream>>>(nodes_from, faces_to,
                                               node_prev, node_pred,
                                               f_nprev, f_cprev, f_npred, f_cpred,
                                               step_p, iter_p, out, n_edges);
}